// Repro_7370163880743
// MI455X (gfx1250) — compile-verified
//
#include <hip/hip_runtime.h>

#define IN_W   456
#define OUT_W  272
#define N_TILE 17            // 17 * 16 = 272 output columns
#define LSTRIDE 460          // padded LDS row stride (floats): 12n mod 64 distinct -> conflict-free
#define SSTRIDE 20           // staging row stride (floats)
#define SCALE_F   1.6764705882352942f
#define INVSUP_F  0.5964912280701754f

typedef __attribute__((ext_vector_type(2))) float v2f;
typedef __attribute__((ext_vector_type(8))) float v8f;

// -------- Setup: banded weight matrix A[17][16][32] into workspace ----------
// Output o (= ct*16 + m) has 5 triangle-filter taps at inputs xmin..xmin+4.
// We place w[j] at column (xmin + j - base(ct)) of A-row m, base = xmin(ct*16) & ~1.
// Band width check: xmin(o0+15)-xmin(o0) <= 26, +1 (even round-down) +4 taps = 31 < 32.
__global__ void resize_weights_kernel(float* __restrict__ A) {
    int o = blockIdx.x * blockDim.x + threadIdx.x;
    if (o >= OUT_W) return;
    float center = ((float)o + 0.5f) * SCALE_F;
    int xmin = (int)(center - SCALE_F + 0.5f); if (xmin < 0) xmin = 0;
    int xmax = (int)(center + SCALE_F + 0.5f); if (xmax > IN_W) xmax = IN_W;
    int size = xmax - xmin; if (size > 5) size = 5;

    float w[5]; float wsum = 0.0f;
#pragma unroll
    for (int j = 0; j < 5; ++j) {
        float t  = ((float)(j + xmin) - center + 0.5f) * INVSUP_F;
        float ww = 1.0f - fminf(fabsf(t), 1.0f);
        ww = (j < size) ? ww : 0.0f;
        w[j] = ww; wsum += ww;
    }
    int ct = o >> 4, m = o & 15;
    // even-aligned band base for this tile (enables 8B-aligned LDS b64 reads)
    float c0 = ((float)(ct << 4) + 0.5f) * SCALE_F;
    int base = (int)(c0 - SCALE_F + 0.5f); if (base < 0) base = 0;
    base &= ~1;

    float* row = A + (ct * 512 + m * 32);
#pragma unroll
    for (int k = 0; k < 32; ++k) row[k] = 0.0f;
#pragma unroll
    for (int j = 0; j < 5; ++j) row[xmin + j - base] = w[j] / wsum;
}

// -------- Main: 16-row stripe per block, banded GEMM via WMMA f32 16x16x4 ---
__launch_bounds__(128)
__global__ void resize_wmma_kernel(const float* __restrict__ in,
                                   const float* __restrict__ A,
                                   float* __restrict__ out) {
    __shared__ float lin[16 * LSTRIDE];       // 29440 B input stripe
    __shared__ float stage[4][16 * SSTRIDE];  // 5120 B  per-wave D transpose

    const int tid = threadIdx.x;
    const int rb  = blockIdx.x << 4;          // first of 16 rows (87552/16 = 5472 blocks)

    // ---- Cooperative load: 16x456 stripe is CONTIGUOUS in global memory ----
    const float4* gin = (const float4*)(in + (size_t)rb * IN_W); // 1824 float4 groups
#pragma unroll
    for (int it = 0; it < 15; ++it) {
        int g = it * 128 + tid;
        if (g < 16 * 114) {                    // 114 float4 per row (456/4)
            int row  = g / 114;
            int col4 = g - row * 114;
            float4 v = gin[g];
            *(float4*)&lin[row * LSTRIDE + (col4 << 2)] = v;
        }
    }
    // zero the pad columns 456..459 (zero-weight taps may read them)
    if (tid < 64) {
        int row = tid >> 2, c = IN_W + (tid & 3);
        lin[row * LSTRIDE + c] = 0.0f;
    }
    __syncthreads();

    const int wave = tid >> 5;
    const int lane = tid & 31;
    const int m    = lane & 15;               // A-row / B-col(N=row) / D lane index
    const int h    = lane >> 4;               // K half-select (K += 2h)
    float* st = &stage[wave][0];

    // each wave owns tiles {wave, wave+4, ...}: EXEC stays all-ones for WMMA
    for (int ct = wave; ct < N_TILE; ct += 4) {
        float c0 = ((float)(ct << 4) + 0.5f) * SCALE_F;
        int base = (int)(c0 - SCALE_F + 0.5f); if (base < 0) base = 0;
        base &= ~1;

        const float* Arow = A + ct * 512 + m * 32;         // L2-resident weights
        const float* Brow = &lin[m * LSTRIDE + base];      // conflict-free b64 reads

        v8f acc = {0.f, 0.f, 0.f, 0.f, 0.f, 0.f, 0.f, 0.f};
#pragma unroll
        for (int c = 0; c < 8; ++c) {                      // K = 32 in 8 steps of 4
            v2f a = *(const v2f*)(Arow + (c << 2) + (h << 1));
            v2f b = *(const v2f*)(Brow + (c << 2) + (h << 1));
            acc = __builtin_amdgcn_wmma_f32_16x16x4_f32(
                false, a, false, b, (short)0, acc, false, false);
        }

        // D layout: lane, vgpr v -> M(outcol) = v + 8h, N(row) = lane&15.
        // Transpose through LDS so the global store is two coalesced b128s.
#pragma unroll
        for (int v = 0; v < 8; ++v)
            st[m * SSTRIDE + v + (h << 3)] = acc[v];
        asm volatile("s_wait_dscnt 0" ::: "memory");       // intra-wave LDS RAW

        int r  = lane >> 1;                                // row 0..15
        int cp = (lane & 1) << 3;                          // col part 0 / 8
        float4 o0v = *(const float4*)&st[r * SSTRIDE + cp];
        float4 o1v = *(const float4*)&st[r * SSTRIDE + cp + 4];
        float* orow = out + (size_t)(rb + r) * OUT_W + (ct << 4) + cp;
        *(float4*)orow       = o0v;
        *(float4*)(orow + 4) = o1v;
        asm volatile("s_wait_dscnt 0" ::: "memory");       // WAR before next tile's stores
    }
}

extern "C" void kernel_launch(void* const* d_in, const int* in_sizes, int n_in,
                              void* d_out, int out_size, void* d_ws, size_t ws_size,
                              hipStream_t stream) {
    const float* in = (const float*)d_in[0];
    float* out = (float*)d_out;
    float* A   = (float*)d_ws;                // needs 17*16*32*4 = 34816 B

    resize_weights_kernel<<<dim3(1), dim3(288), 0, stream>>>(A);

    const int rows = 64 * 3 * 456;            // 87552, divisible by 16
    resize_wmma_kernel<<<dim3(rows / 16), dim3(128), 0, stream>>>(in, A, out);
    (void)in_sizes; (void)n_in; (void)out_size; (void)ws_size;
}